// QuanvolutionFilter_65481071395904
// MI455X (gfx1250) — compile-verified
//
#include <hip/hip_runtime.h>
#include <hip/hip_bf16.h>

typedef __attribute__((ext_vector_type(2))) float v2f;
typedef __attribute__((ext_vector_type(8))) float v8f;

// ---------------------------------------------------------------------------
// Setup: build W = U^T (16x16) where U is the fixed variational circuit.
// Thread k simulates the circuit applied to basis state |k> and writes
// W[k*16 + n] = U[n][k]. Native __cosf/__sinf: half-angles are small
// (params ~ N(0,1)), so v_cos_f32/v_sin_f32 precision is ample, and it
// avoids pulling in OCML's Payne-Hanek slow path (thousands of SALU ops).
// Index convention: basis index i has qubit w at bit (3-w).
// ---------------------------------------------------------------------------
__global__ void quanv_build_W(const float* __restrict__ params,
                              float* __restrict__ W) {
    const int k = threadIdx.x;   // 0..15
    float st[16];
#pragma unroll
    for (int i = 0; i < 16; ++i) st[i] = (i == k) ? 1.0f : 0.0f;

#pragma unroll
    for (int l = 0; l < 2; ++l) {
        // RY(params[l][w]) on each wire
#pragma unroll
        for (int w = 0; w < 4; ++w) {
            const float th = 0.5f * params[l * 4 + w];
            const float c = __cosf(th);
            const float s = __sinf(th);
            const int mask = 1 << (3 - w);
#pragma unroll
            for (int i = 0; i < 16; ++i) {
                if (!(i & mask)) {
                    const float v0 = st[i];
                    const float v1 = st[i | mask];
                    st[i]        = c * v0 - s * v1;   // matches reference RY
                    st[i | mask] = s * v0 + c * v1;
                }
            }
        }
        // CNOT ring: ctrl=w, tgt=(w+1)%4 : swap target pair in ctrl=1 subspace
#pragma unroll
        for (int w = 0; w < 4; ++w) {
            const int cm = 1 << (3 - w);
            const int tm = 1 << (3 - ((w + 1) & 3));
#pragma unroll
            for (int i = 0; i < 16; ++i) {
                if ((i & cm) && !(i & tm)) {
                    const float t = st[i];
                    st[i] = st[i | tm];
                    st[i | tm] = t;
                }
            }
        }
    }
#pragma unroll
    for (int n = 0; n < 16; ++n) W[k * 16 + n] = st[n];
}

// ---------------------------------------------------------------------------
// One Walsh-Hadamard butterfly stage via ds_swizzle_b32 (group-of-32 XOR
// pattern, single DS op, no index VGPRs / bounds checks).
// offset encoding: [14:10]=xor_mask, [9:5]=or_mask=0, [4:0]=and_mask=0x1f.
// ---------------------------------------------------------------------------
template <int XORBIT>
__device__ __forceinline__ float wht_stage(float v, int lane) {
    constexpr int IMM = (XORBIT << 10) | 0x1f;
    const float t = __int_as_float(
        __builtin_amdgcn_ds_swizzle(__float_as_int(v), IMM));
    return (lane & XORBIT) ? (t - v) : (v + t);
}

// ---------------------------------------------------------------------------
// Main kernel: one wave32 handles 16 consecutive patches.
//   A (16 patches x 16 amps, fp32 A-layout) x B (W = U^T) via 4 chained
//   V_WMMA_F32_16X16X4_F32, then square + 16-lane Walsh-Hadamard butterfly
//   to extract the four single-bit coefficients z0..z3 per patch.
// ---------------------------------------------------------------------------
#define TPB 256
#define WAVES_PER_BLOCK (TPB / 32)

__global__ void __launch_bounds__(TPB)
quanv_main(const float* __restrict__ x, const float* __restrict__ W,
           float* __restrict__ out, int n_tiles) {
    __shared__ float lds[WAVES_PER_BLOCK * 64];

    const int lane      = threadIdx.x & 31;
    const int waveInBlk = threadIdx.x >> 5;
    const int tile      = blockIdx.x * WAVES_PER_BLOCK + waveInBlk;
    if (tile >= n_tiles) return;           // uniform per wave (EXEC stays all-1)

    const int h = lane >> 4;               // K half-select (0/1)
    const int m = lane & 15;               // patch-in-tile == M == N index
    const int n = tile * 16 + m;           // global patch id

    // n = b*196 + pr*14 + pc ; pixels at x[b, 2pr + {0,1}, 2pc + {0,1}]
    const int b  = n / 196;
    const int p  = n - b * 196;
    const int pr = p / 14;
    const int pc = p - pr * 14;
    const float* px = x + b * 784 + (pr * 2) * 28 + (pc * 2);
    const float2 r0 = *reinterpret_cast<const float2*>(px);        // (r,c),(r,c+1)
    const float2 r1 = *reinterpret_cast<const float2*>(px + 28);   // (r+1,c),(r+1,c+1)

    // half-angle cos/sin per wire (wire order: r0.x, r0.y, r1.x, r1.y)
    const float c0 = __cosf(0.5f * r0.x), s0 = __sinf(0.5f * r0.x);
    const float c1 = __cosf(0.5f * r0.y), s1 = __sinf(0.5f * r0.y);
    const float c2 = __cosf(0.5f * r1.x), s2 = __sinf(0.5f * r1.x);
    const float c3 = __cosf(0.5f * r1.y), s3 = __sinf(0.5f * r1.y);

    // Amplitude psi_k for k = 4c + 2h + r  (bit3=c>>1, bit2=c&1, bit1=h, bit0=r)
    const float t2 = h ? s2 : c2;
    const float f0 = t2 * c3;
    const float f1 = t2 * s3;

    float a[8], bb[8];
#pragma unroll
    for (int c = 0; c < 4; ++c) {
        const float t01 = ((c & 2) ? s0 : c0) * ((c & 1) ? s1 : c1);
        a[2 * c]     = t01 * f0;
        a[2 * c + 1] = t01 * f1;
        const int k0 = 4 * c + 2 * h;
        bb[2 * c]     = W[k0 * 16 + m];        // B[k][n] = U[n][k], coalesced in m
        bb[2 * c + 1] = W[(k0 + 1) * 16 + m];
    }

    // Phi = Psi * U^T : 4 chained fp32 WMMAs over K = 16
    v8f acc = {};
#pragma unroll
    for (int c = 0; c < 4; ++c) {
        v2f av; av[0] = a[2 * c];  av[1] = a[2 * c + 1];
        v2f bv; bv[0] = bb[2 * c]; bv[1] = bb[2 * c + 1];
        acc = __builtin_amdgcn_wmma_f32_16x16x4_f32(
            false, av, false, bv, (short)0, acc, false, false);
    }

    // Square and 16-lane WHT butterfly: after 4 stages, lane j (per 16-half)
    // holds sum_i (-1)^popcount(i&j) * p_i ; single-bit j = 2^(3-w) gives z_w.
    float z[8];
#pragma unroll
    for (int r = 0; r < 8; ++r) {
        float v = acc[r] * acc[r];
        v = wht_stage<1>(v, lane);
        v = wht_stage<2>(v, lane);
        v = wht_stage<4>(v, lane);
        v = wht_stage<8>(v, lane);
        z[r] = v;
    }

    // Scatter z into LDS: VGPR r holds patch M = r + 8*h; lane j in {1,2,4,8}
    // carries wire w = 4 - ffs(j).
    float* wl = lds + waveInBlk * 64;
    const int j = lane & 15;
    if (j && !(j & (j - 1))) {
        const int w = 4 - __ffs(j);
#pragma unroll
        for (int r = 0; r < 8; ++r)
            wl[(r + h * 8) * 4 + w] = z[r];
    }
    // Same-wave LDS RAW: DS ops are in-order; wait + compiler fence.
    asm volatile("s_wait_dscnt 0x0" ::: "memory");

    // Coalesced output: tile covers out[tile*64 .. tile*64+63]
    const float2 o = *reinterpret_cast<const float2*>(wl + lane * 2);
    *reinterpret_cast<float2*>(out + tile * 64 + lane * 2) = o;
}

// ---------------------------------------------------------------------------
extern "C" void kernel_launch(void* const* d_in, const int* in_sizes, int n_in,
                              void* d_out, int out_size, void* d_ws, size_t ws_size,
                              hipStream_t stream) {
    const float* x      = (const float*)d_in[0];   // (B, 28, 28) f32
    const float* params = (const float*)d_in[1];   // (2, 4) f32
    float* out = (float*)d_out;                    // (B, 784) f32
    float* W   = (float*)d_ws;                     // 16x16 f32 scratch

    hipLaunchKernelGGL(quanv_build_W, dim3(1), dim3(16), 0, stream, params, W);

    const int B        = in_sizes[0] / 784;        // 8192
    const int nPatches = B * 196;                  // 1,605,632 (multiple of 16)
    const int nTiles   = nPatches / 16;            // 100,352
    const int blocks   = (nTiles + WAVES_PER_BLOCK - 1) / WAVES_PER_BLOCK;

    hipLaunchKernelGGL(quanv_main, dim3(blocks), dim3(TPB), 0, stream,
                       x, W, out, nTiles);
}